// SimpleGNN_39058432589890
// MI455X (gfx1250) — compile-verified
//
#include <hip/hip_runtime.h>
#include <math.h>

// ---------------------------------------------------------------------------
// 2-layer GCN (PyG GCNConv x2 + log_softmax) for MI455X / gfx1250.
//   N=50000 nodes, E=800000 edges, 64 -> 64 -> 32 features, fp32.
// Bottleneck is the edge scatter-add (~0.6 GB of L2-resident traffic);
// GEMMs use V_WMMA_F32_16X16X4_F32 (fp32 matrix path, full accuracy).
// ---------------------------------------------------------------------------

typedef float v2f __attribute__((ext_vector_type(2)));
typedef float v8f __attribute__((ext_vector_type(8)));

#define WMMA_F32(A, B, C) \
  __builtin_amdgcn_wmma_f32_16x16x4_f32(false, (A), false, (B), (short)0, (C), false, false)

// ---------------- degree / normalization ----------------------------------

__global__ __launch_bounds__(256) void deg_init_kernel(float* __restrict__ deg, int n) {
  int i = blockIdx.x * 256 + threadIdx.x;
  if (i < n) deg[i] = 1.0f;  // self-loop
}

__global__ __launch_bounds__(256) void deg_count_kernel(const int* __restrict__ dst,
                                                        float* __restrict__ deg, int E) {
  int e = blockIdx.x * 256 + threadIdx.x;
  if (e < E) unsafeAtomicAdd(&deg[dst[e]], 1.0f);
}

__global__ __launch_bounds__(256) void deg_rsqrt_kernel(float* __restrict__ deg, int n) {
  int i = blockIdx.x * 256 + threadIdx.x;
  if (i < n) deg[i] = rsqrtf(deg[i]);  // in-place: deg -> dinv
}

// ---------------- layer 1 GEMM: h1 = x @ W1 ; agg1 = h1 * dinv^2 ----------
// 128 threads = 4 waves; each wave does a 16-row strip across all 64 cols.

__global__ __launch_bounds__(128) void gemm1_kernel(const float* __restrict__ x,
                                                    const float* __restrict__ W1,
                                                    const float* __restrict__ dinv,
                                                    float* __restrict__ h1,
                                                    float* __restrict__ agg1, int n) {
  __shared__ float lw[64 * 64];
  for (int i = threadIdx.x; i < 64 * 64; i += 128) lw[i] = W1[i];
  __syncthreads();

  const int wave = threadIdx.x >> 5;
  const int lane = threadIdx.x & 31;
  const int m    = lane & 15;   // A row / B,C col within tile
  const int kh   = lane >> 4;   // K half select
  const int rowBase = blockIdx.x * 64 + wave * 16;
  if (rowBase >= n) return;     // wave-uniform: EXEC stays all-1s for WMMA

  v8f acc0 = {}, acc1 = {}, acc2 = {}, acc3 = {};
  const float* xrow = x + (size_t)(rowBase + m) * 64;

#pragma unroll
  for (int kk = 0; kk < 16; ++kk) {
    const int k0 = kk * 4 + kh * 2;  // K = k0, k0+1 for regs 0,1 of this lane
    v2f a;
    a.x = xrow[k0];
    a.y = xrow[k0 + 1];
    v2f b0, b1v, b2v, b3;
    b0.x  = lw[k0 * 64 + m];            b0.y  = lw[(k0 + 1) * 64 + m];
    b1v.x = lw[k0 * 64 + 16 + m];       b1v.y = lw[(k0 + 1) * 64 + 16 + m];
    b2v.x = lw[k0 * 64 + 32 + m];       b2v.y = lw[(k0 + 1) * 64 + 32 + m];
    b3.x  = lw[k0 * 64 + 48 + m];       b3.y  = lw[(k0 + 1) * 64 + 48 + m];
    acc0 = WMMA_F32(a, b0, acc0);
    acc1 = WMMA_F32(a, b1v, acc1);
    acc2 = WMMA_F32(a, b2v, acc2);
    acc3 = WMMA_F32(a, b3, acc3);
  }

#pragma unroll
  for (int r = 0; r < 8; ++r) {
    const int row = rowBase + r + 8 * kh;
    const float di = dinv[row];
    const float w  = di * di;            // self-loop norm
    const size_t base = (size_t)row * 64 + m;
    h1[base +  0] = acc0[r];  agg1[base +  0] = acc0[r] * w;
    h1[base + 16] = acc1[r];  agg1[base + 16] = acc1[r] * w;
    h1[base + 32] = acc2[r];  agg1[base + 32] = acc2[r] * w;
    h1[base + 48] = acc3[r];  agg1[base + 48] = acc3[r] * w;
  }
}

// ---------------- edge scatter-add, 64-wide -------------------------------
// One thread per (edge, feature). Atomics resolve in L2 (h1 is L2-resident).

__global__ __launch_bounds__(256) void agg64_kernel(const int* __restrict__ src,
                                                    const int* __restrict__ dst,
                                                    const float* __restrict__ dinv,
                                                    const float* __restrict__ h,
                                                    float* __restrict__ agg, int E) {
  long tid = (long)blockIdx.x * 256 + threadIdx.x;
  if (tid >= (long)E * 64) return;
  const int e = (int)(tid >> 6);
  const int f = (int)(tid & 63);
  const int s = src[e], d = dst[e];
  const float w = dinv[s] * dinv[d];
  unsafeAtomicAdd(&agg[(size_t)d * 64 + f], h[(size_t)s * 64 + f] * w);
}

// ---------------- layer 2 GEMM: h2 = relu(agg1 + b1) @ W2 -----------------

__global__ __launch_bounds__(128) void gemm2_kernel(const float* __restrict__ agg1,
                                                    const float* __restrict__ W2,
                                                    const float* __restrict__ b1,
                                                    const float* __restrict__ dinv,
                                                    float* __restrict__ h2,
                                                    float* __restrict__ agg2, int n) {
  __shared__ float lw[64 * 32];
  __shared__ float lb[64];
  for (int i = threadIdx.x; i < 64 * 32; i += 128) lw[i] = W2[i];
  if (threadIdx.x < 64) lb[threadIdx.x] = b1[threadIdx.x];
  __syncthreads();

  const int wave = threadIdx.x >> 5;
  const int lane = threadIdx.x & 31;
  const int m    = lane & 15;
  const int kh   = lane >> 4;
  const int rowBase = blockIdx.x * 64 + wave * 16;
  if (rowBase >= n) return;

  v8f acc0 = {}, acc1 = {};
  const float* arow = agg1 + (size_t)(rowBase + m) * 64;

#pragma unroll
  for (int kk = 0; kk < 16; ++kk) {
    const int k0 = kk * 4 + kh * 2;
    float a0 = arow[k0]     + lb[k0];
    float a1 = arow[k0 + 1] + lb[k0 + 1];
    v2f a;
    a.x = a0 > 0.0f ? a0 : 0.0f;   // fused ReLU of layer-1 output
    a.y = a1 > 0.0f ? a1 : 0.0f;
    v2f b0, b1v;
    b0.x  = lw[k0 * 32 + m];        b0.y  = lw[(k0 + 1) * 32 + m];
    b1v.x = lw[k0 * 32 + 16 + m];   b1v.y = lw[(k0 + 1) * 32 + 16 + m];
    acc0 = WMMA_F32(a, b0, acc0);
    acc1 = WMMA_F32(a, b1v, acc1);
  }

#pragma unroll
  for (int r = 0; r < 8; ++r) {
    const int row = rowBase + r + 8 * kh;
    const float di = dinv[row];
    const float w  = di * di;
    const size_t base = (size_t)row * 32 + m;
    h2[base +  0] = acc0[r];  agg2[base +  0] = acc0[r] * w;
    h2[base + 16] = acc1[r];  agg2[base + 16] = acc1[r] * w;
  }
}

// ---------------- edge scatter-add, 32-wide -------------------------------

__global__ __launch_bounds__(256) void agg32_kernel(const int* __restrict__ src,
                                                    const int* __restrict__ dst,
                                                    const float* __restrict__ dinv,
                                                    const float* __restrict__ h,
                                                    float* __restrict__ agg, int E) {
  long tid = (long)blockIdx.x * 256 + threadIdx.x;
  if (tid >= (long)E * 32) return;
  const int e = (int)(tid >> 5);
  const int f = (int)(tid & 31);
  const int s = src[e], d = dst[e];
  const float w = dinv[s] * dinv[d];
  unsafeAtomicAdd(&agg[(size_t)d * 32 + f], h[(size_t)s * 32 + f] * w);
}

// ---------------- log_softmax over 32 classes (one wave per row) ----------

__global__ __launch_bounds__(128) void logsoftmax_kernel(const float* __restrict__ agg2,
                                                         const float* __restrict__ b2,
                                                         float* __restrict__ out, int n) {
  const int row = blockIdx.x * 4 + (threadIdx.x >> 5);
  const int c   = threadIdx.x & 31;
  if (row >= n) return;
  float z = agg2[(size_t)row * 32 + c] + b2[c];
  float mx = z;
#pragma unroll
  for (int off = 16; off > 0; off >>= 1) mx = fmaxf(mx, __shfl_xor(mx, off, 32));
  float e = expf(z - mx);
  float s = e;
#pragma unroll
  for (int off = 16; off > 0; off >>= 1) s += __shfl_xor(s, off, 32);
  out[(size_t)row * 32 + c] = (z - mx) - logf(s);
}

// ---------------------------------------------------------------------------

extern "C" void kernel_launch(void* const* d_in, const int* in_sizes, int n_in,
                              void* d_out, int out_size, void* d_ws, size_t ws_size,
                              hipStream_t stream) {
  const float* x  = (const float*)d_in[0];
  const int*   ei = (const int*)d_in[1];
  const float* W1 = (const float*)d_in[2];
  const float* b1 = (const float*)d_in[3];
  const float* W2 = (const float*)d_in[4];
  const float* b2 = (const float*)d_in[5];

  const int N = in_sizes[0] / 64;
  const int E = in_sizes[1] / 2;
  const int* src = ei;
  const int* dst = ei + E;

  // workspace (floats): dinv[N] | hbuf[64N] (h1, reused as h2) | agg1[64N] | agg2[32N]
  float* ws   = (float*)d_ws;
  float* dinv = ws;                          // N  (deg -> rsqrt in place)
  float* hbuf = dinv + N;                    // 64N
  float* agg1 = hbuf + (size_t)N * 64;       // 64N
  float* agg2 = agg1 + (size_t)N * 64;       // 32N
  float* out  = (float*)d_out;

  const int TB = 256;
  // 1. degrees of (A + I) counted on dst, then dinv = rsqrt(deg)
  deg_init_kernel<<<(N + TB - 1) / TB, TB, 0, stream>>>(dinv, N);
  deg_count_kernel<<<(E + TB - 1) / TB, TB, 0, stream>>>(dst, dinv, E);
  deg_rsqrt_kernel<<<(N + TB - 1) / TB, TB, 0, stream>>>(dinv, N);

  // 2. h1 = x @ W1 (WMMA f32); agg1 initialized with self-loop term h1*dinv^2
  gemm1_kernel<<<(N + 63) / 64, 128, 0, stream>>>(x, W1, dinv, hbuf, agg1, N);

  // 3. agg1[d] += h1[s] * dinv[s]*dinv[d] over edges (64-wide)
  {
    long total = (long)E * 64;
    agg64_kernel<<<(unsigned)((total + TB - 1) / TB), TB, 0, stream>>>(src, dst, dinv, hbuf, agg1, E);
  }

  // 4. h2 = relu(agg1 + b1) @ W2 (WMMA f32); agg2 init with self-loop term
  gemm2_kernel<<<(N + 63) / 64, 128, 0, stream>>>(agg1, W2, b1, dinv, hbuf, agg2, N);

  // 5. agg2[d] += h2[s] * dinv[s]*dinv[d] over edges (32-wide)
  {
    long total = (long)E * 32;
    agg32_kernel<<<(unsigned)((total + TB - 1) / TB), TB, 0, stream>>>(src, dst, dinv, hbuf, agg2, E);
  }

  // 6. out = log_softmax(agg2 + b2, axis=1)
  logsoftmax_kernel<<<(N + 3) / 4, 128, 0, stream>>>(agg2, b2, out, N);
}